// TritonFFTLarge_53584011985645
// MI455X (gfx1250) — compile-verified
//
#include <hip/hip_runtime.h>

typedef __attribute__((ext_vector_type(2))) float v2f;
typedef __attribute__((ext_vector_type(8))) float v8f;

#define WAVES 4
#define TWO_PI 6.283185307179586f

static __device__ __forceinline__ v8f wmma_f32(v2f a, v2f b, v8f c) {
  // D = A(16x4) x B(4x16) + C, full fp32
  return __builtin_amdgcn_wmma_f32_16x16x4_f32(false, a, false, b, (short)0, c,
                                               false, false);
}

__global__ void __launch_bounds__(WAVES * 32)
fft1024_wmma(const float* __restrict__ xre_g, const float* __restrict__ xim_g,
             float* __restrict__ yre_g, float* __restrict__ yim_g, int nfft) {
  // per-wave 32x33 padded tile of (re,im) pairs: 4*32*33*8 = 33,792 B
  __shared__ v2f lds[WAVES][32][33];

  const int lane = (int)(threadIdx.x & 31u);
  const int wave = (int)(threadIdx.x >> 5);
  const int fft  = (int)blockIdx.x * WAVES + wave;
  if (fft >= nfft) return;  // wave-uniform: EXEC stays all-ones below

  const int l15 = lane & 15;
  const int hi  = lane >> 4;  // lane half (K/M split in WMMA layouts)

  // ---- DFT-32 matrix as A-operand fragments (symmetric: reused both stages)
  // A 16x4 layout: lane = M (0..15), vgpr v + 2*hi = K
  v2f Fr[2][8], Fi[2][8];
#pragma unroll
  for (int mi = 0; mi < 2; ++mi) {
#pragma unroll
    for (int kk = 0; kk < 8; ++kk) {
      int m = mi * 16 + l15;
#pragma unroll
      for (int v = 0; v < 2; ++v) {
        int k = kk * 4 + 2 * hi + v;
        float ang = (float)((m * k) & 31) * (-TWO_PI / 32.0f);
        Fr[mi][kk][v] = __cosf(ang);
        Fi[mi][kk][v] = __sinf(ang);
      }
    }
  }

  const float* xr = xre_g + (size_t)fft * 1024;
  const float* xi = xim_g + (size_t)fft * 1024;

  // ---- load X (32x32, X[n1][n2] = x[n1*32+n2]) as B-operand fragments
  // B 4x16 layout (mirror of A): lane = N, vgpr v + 2*hi = K
  v2f Xr[8][2], Xi[8][2];
#pragma unroll
  for (int kk = 0; kk < 8; ++kk) {
#pragma unroll
    for (int nj = 0; nj < 2; ++nj) {
#pragma unroll
      for (int v = 0; v < 2; ++v) {
        int idx = (kk * 4 + 2 * hi + v) * 32 + nj * 16 + l15;  // n1*32 + n2
        Xr[kk][nj][v] = xr[idx];
        Xi[kk][nj][v] = xi[idx];
      }
    }
  }

  // ---- stage 1: B1 = F32 x X (complex), then twiddle, then LDS (row-major)
#pragma unroll
  for (int mi = 0; mi < 2; ++mi) {
#pragma unroll
    for (int nj = 0; nj < 2; ++nj) {
      v8f aR = {}, aT = {}, aI = {};
#pragma unroll
      for (int kk = 0; kk < 8; ++kk) {
        aR = wmma_f32(Fr[mi][kk], Xr[kk][nj], aR);  // Re+ : Fr*Xr
        aT = wmma_f32(Fi[mi][kk], Xi[kk][nj], aT);  // Re- : Fi*Xi
        aI = wmma_f32(Fr[mi][kk], Xi[kk][nj], aI);  // Im  : Fr*Xi
        aI = wmma_f32(Fi[mi][kk], Xr[kk][nj], aI);  // Im  : Fi*Xr
      }
      // C/D layout: lane half -> M+8, vgpr r -> M; lane l15 -> N
#pragma unroll
      for (int r = 0; r < 8; ++r) {
        float br = aR[r] - aT[r];
        float bi = aI[r];
        int k1 = mi * 16 + 8 * hi + r;
        int n2 = nj * 16 + l15;
        float ang = (float)(k1 * n2) * (-TWO_PI / 1024.0f);  // k1*n2 <= 961
        float c = __cosf(ang), s = __sinf(ang);
        v2f cv;
        cv.x = br * c - bi * s;
        cv.y = br * s + bi * c;
        lds[wave][k1][n2] = cv;  // C[k1][n2]
      }
    }
  }

  // ---- gather C^T as B-operand fragments (transposed LDS read)
  // frag(kk,nj): K = n2 = kk*4 + v + 2*hi, N = k1 = nj*16 + l15
  v2f CtR[8][2], CtI[8][2];
#pragma unroll
  for (int kk = 0; kk < 8; ++kk) {
#pragma unroll
    for (int nj = 0; nj < 2; ++nj) {
      int k1 = nj * 16 + l15;
      int n2 = kk * 4 + 2 * hi;
      v2f e0 = lds[wave][k1][n2];
      v2f e1 = lds[wave][k1][n2 + 1];
      v2f rr; rr.x = e0.x; rr.y = e1.x;
      v2f ii; ii.x = e0.y; ii.y = e1.y;
      CtR[kk][nj] = rr;
      CtI[kk][nj] = ii;
    }
  }

  float* yr = yre_g + (size_t)fft * 1024;
  float* yi = yim_g + (size_t)fft * 1024;

  // ---- stage 3: Y[k2][k1] = F32 x C^T ; flat out index = k2*32 + k1 (coalesced)
#pragma unroll
  for (int mi = 0; mi < 2; ++mi) {
#pragma unroll
    for (int nj = 0; nj < 2; ++nj) {
      v8f aR = {}, aT = {}, aI = {};
#pragma unroll
      for (int kk = 0; kk < 8; ++kk) {
        aR = wmma_f32(Fr[mi][kk], CtR[kk][nj], aR);
        aT = wmma_f32(Fi[mi][kk], CtI[kk][nj], aT);
        aI = wmma_f32(Fr[mi][kk], CtI[kk][nj], aI);
        aI = wmma_f32(Fi[mi][kk], CtR[kk][nj], aI);
      }
#pragma unroll
      for (int r = 0; r < 8; ++r) {
        int k2 = mi * 16 + 8 * hi + r;
        int idx = k2 * 32 + nj * 16 + l15;
        yr[idx] = aR[r] - aT[r];
        yi[idx] = aI[r];
      }
    }
  }
}

extern "C" void kernel_launch(void* const* d_in, const int* in_sizes, int n_in,
                              void* d_out, int out_size, void* d_ws,
                              size_t ws_size, hipStream_t stream) {
  (void)n_in; (void)out_size; (void)d_ws; (void)ws_size;
  const float* xre = (const float*)d_in[0];
  const float* xim = (const float*)d_in[1];
  const int n_total = in_sizes[0];        // batch * 1024
  const int nfft    = n_total / 1024;
  float* yre = (float*)d_out;             // tuple (re, im) concatenated flat
  float* yim = (float*)d_out + (size_t)n_total;
  const int blocks = (nfft + WAVES - 1) / WAVES;
  fft1024_wmma<<<blocks, WAVES * 32, 0, stream>>>(xre, xim, yre, yim, nfft);
}